// MPTAttention_24206435680858
// MI455X (gfx1250) — compile-verified
//
#include <hip/hip_runtime.h>
#include <hip/hip_bf16.h>
#include <stdint.h>

// ---------------------------------------------------------------------------
// MPT attention layer for MI455X (gfx1250, wave32, WMMA).
// Memory-bound workload (~100 GFLOP vs ~200MB data) -> use full-precision
// V_WMMA_F32_16X16X4_F32 everywhere; no benefit from bf16 downconversion.
// Round 2: stage V transposed in LDS (kills the per-WMMA global-load waits
// seen in round 1 asm), deepen GEMM K-stage to 32 to halve barriers.
// ---------------------------------------------------------------------------

typedef __attribute__((ext_vector_type(2))) float v2f;
typedef __attribute__((ext_vector_type(8))) float v8f;

#define T_TOK   2048
#define D_MODEL 2048
#define N_QKV   6144
#define H_HEADS 16
#define HD      128
#define K_SK    512
#define CLIP_V  8.0f
#define LN_EPS  1e-5f
#define WCA_EPS 1e-6f
#define SM_SCALE 0.08838834764831845f   // 1/sqrt(128)

__device__ __forceinline__ v8f wmma_f32(v2f a, v2f b, v8f c) {
    // D = A(16x4,f32) x B(4x16,f32) + C(16x16,f32)
    return __builtin_amdgcn_wmma_f32_16x16x4_f32(
        false, a, false, b, (short)0, c, false, false);
}

// in-order LDS + compiler reordering fence (per-wave phase transition)
#define LDS_FENCE() asm volatile("s_wait_dscnt 0x0" ::: "memory")

// ---------------------------------------------------------------------------
// Kernel 1/5: NT GEMM  C[m][n] = sum_k A[m][k] * B[n][k]   (both row-major)
// 64x64 tile / 128 threads (4 waves), each wave computes a 32x32 quadrant.
// K staged 32 deep per barrier pair; stride-36 rows: 16B aligned, no bank
// conflicts ((36*r+c)%64 distinct over 16 consecutive rows).
// ---------------------------------------------------------------------------
template<bool DO_CLIP>
__global__ __launch_bounds__(128)
void gemm_nt_kernel(const float* __restrict__ A, const float* __restrict__ B,
                    float* __restrict__ C, int M, int N, int K)
{
    __shared__ float As[64][36];
    __shared__ float Bs[64][36];

    const int tid  = threadIdx.x;
    const int lane = tid & 31;
    const int wave = tid >> 5;           // 0..3
    const int lrow = lane & 15;
    const int lhi  = lane >> 4;          // 0/1

    const int mTile = blockIdx.y * 64;
    const int nTile = blockIdx.x * 64;
    const int wRow  = (wave >> 1) * 32;  // 0 / 32
    const int wCol  = (wave & 1) * 32;

    const int sRow = tid >> 1;           // staging: 0..63
    const int sCol = (tid & 1) * 16;     // 0 / 16

    v8f acc[2][2] = {};

    for (int kb = 0; kb < K; kb += 32) {
        const float* ap = A + (size_t)(mTile + sRow) * K + kb + sCol;
        const float* bp = B + (size_t)(nTile + sRow) * K + kb + sCol;
        float4 a0 = *(const float4*)(ap);
        float4 a1 = *(const float4*)(ap + 4);
        float4 a2 = *(const float4*)(ap + 8);
        float4 a3 = *(const float4*)(ap + 12);
        float4 b0 = *(const float4*)(bp);
        float4 b1 = *(const float4*)(bp + 4);
        float4 b2 = *(const float4*)(bp + 8);
        float4 b3 = *(const float4*)(bp + 12);
        __syncthreads();
        *(float4*)&As[sRow][sCol]      = a0;
        *(float4*)&As[sRow][sCol + 4]  = a1;
        *(float4*)&As[sRow][sCol + 8]  = a2;
        *(float4*)&As[sRow][sCol + 12] = a3;
        *(float4*)&Bs[sRow][sCol]      = b0;
        *(float4*)&Bs[sRow][sCol + 4]  = b1;
        *(float4*)&Bs[sRow][sCol + 8]  = b2;
        *(float4*)&Bs[sRow][sCol + 12] = b3;
        __syncthreads();

        #pragma unroll
        for (int kk = 0; kk < 32; kk += 4) {
            v2f aF0 = *(const v2f*)&As[wRow +      lrow][kk + 2 * lhi];
            v2f aF1 = *(const v2f*)&As[wRow + 16 + lrow][kk + 2 * lhi];
            v2f bF0 = *(const v2f*)&Bs[wCol +      lrow][kk + 2 * lhi];
            v2f bF1 = *(const v2f*)&Bs[wCol + 16 + lrow][kk + 2 * lhi];
            acc[0][0] = wmma_f32(aF0, bF0, acc[0][0]);
            acc[0][1] = wmma_f32(aF0, bF1, acc[0][1]);
            acc[1][0] = wmma_f32(aF1, bF0, acc[1][0]);
            acc[1][1] = wmma_f32(aF1, bF1, acc[1][1]);
        }
    }

    #pragma unroll
    for (int rt = 0; rt < 2; ++rt)
        #pragma unroll
        for (int ct = 0; ct < 2; ++ct)
            #pragma unroll
            for (int r = 0; r < 8; ++r) {
                int gm = mTile + wRow + rt * 16 + r + 8 * lhi;
                int gn = nTile + wCol + ct * 16 + lrow;
                float v = acc[rt][ct][r];
                if (DO_CLIP) v = fminf(fmaxf(v, -CLIP_V), CLIP_V);
                C[(size_t)gm * N + gn] = v;
            }
}

// ---------------------------------------------------------------------------
// Kernel 2: LayerNorm in-place on q / k slices of qkv [T][6144]
// blockIdx.x = row*2 + (0:q, 1:k)
// ---------------------------------------------------------------------------
__global__ __launch_bounds__(256)
void ln_kernel(float* __restrict__ qkv,
               const float* __restrict__ qw, const float* __restrict__ qb,
               const float* __restrict__ kw, const float* __restrict__ kb)
{
    const int row   = blockIdx.x >> 1;
    const int which = blockIdx.x & 1;
    float* x = qkv + (size_t)row * N_QKV + which * D_MODEL;
    const float* w = which ? kw : qw;
    const float* b = which ? kb : qb;
    const int tid = threadIdx.x;

    float v[8];
    float s = 0.f, sq = 0.f;
    #pragma unroll
    for (int i = 0; i < 8; ++i) {
        float t = x[tid + 256 * i];
        v[i] = t; s += t; sq += t * t;
    }
    __shared__ float rs[256], rq[256];
    rs[tid] = s; rq[tid] = sq;
    __syncthreads();
    for (int off = 128; off; off >>= 1) {
        if (tid < off) { rs[tid] += rs[tid + off]; rq[tid] += rq[tid + off]; }
        __syncthreads();
    }
    float mean = rs[0] * (1.0f / D_MODEL);
    float var  = rq[0] * (1.0f / D_MODEL) - mean * mean;
    float inv  = rsqrtf(var + LN_EPS);
    #pragma unroll
    for (int i = 0; i < 8; ++i) {
        int c = tid + 256 * i;
        x[c] = (v[i] - mean) * inv * w[c] + b[c];
    }
}

// ---------------------------------------------------------------------------
// Kernel 3: cache-reuse gather -> compact kf/vf [T][D]
// ---------------------------------------------------------------------------
__global__ __launch_bounds__(256)
void gather_kernel(const float* __restrict__ qkv,
                   const float* __restrict__ kcache, const float* __restrict__ vcache,
                   const int* __restrict__ cache_idx,
                   float* __restrict__ kf, float* __restrict__ vf)
{
    const int t   = blockIdx.x;
    const int idx = cache_idx[t];
    const bool reuse = idx >= 0;
    const int ci = reuse ? idx : 0;
    #pragma unroll
    for (int i = 0; i < 8; ++i) {
        int d = threadIdx.x + 256 * i;
        float kk = reuse ? kcache[(size_t)ci * D_MODEL + d]
                         : qkv[(size_t)t * N_QKV + D_MODEL + d];
        float vv = reuse ? vcache[(size_t)ci * D_MODEL + d]
                         : qkv[(size_t)t * N_QKV + 2 * D_MODEL + d];
        kf[(size_t)t * D_MODEL + d] = kk;
        vf[(size_t)t * D_MODEL + d] = vv;
    }
}

// ---------------------------------------------------------------------------
// Kernel 4: alpha blend at Sk rows (unique indices -> race free)
// ---------------------------------------------------------------------------
__global__ __launch_bounds__(256)
void blend_kernel(const float* __restrict__ qkv,
                  const float* __restrict__ kcache, const float* __restrict__ vcache,
                  const int* __restrict__ cache_idx, const int* __restrict__ Sk,
                  float* __restrict__ kf, float* __restrict__ vf)
{
    const int t = Sk[blockIdx.x];
    int idx = cache_idx[t]; if (idx < 0) idx = 0;
    const int tid = threadIdx.x;

    float kr[8], krec[8];
    float num = 0.f, den = 0.f;
    #pragma unroll
    for (int i = 0; i < 8; ++i) {
        int d = tid + 256 * i;
        kr[i]   = kcache[(size_t)idx * D_MODEL + d];
        krec[i] = qkv[(size_t)t * N_QKV + D_MODEL + d];
        float df = krec[i] - kr[i];
        num += df * df;
        den += kr[i] * kr[i];
    }
    __shared__ float rn[256], rd[256];
    rn[tid] = num; rd[tid] = den;
    __syncthreads();
    for (int off = 128; off; off >>= 1) {
        if (tid < off) { rn[tid] += rn[tid + off]; rd[tid] += rd[tid + off]; }
        __syncthreads();
    }
    float alpha = rn[0] / fmaxf(rd[0], WCA_EPS);
    alpha = fminf(fmaxf(alpha, 0.f), 1.f);
    #pragma unroll
    for (int i = 0; i < 8; ++i) {
        int d = tid + 256 * i;
        kf[(size_t)t * D_MODEL + d] = alpha * krec[i] + (1.f - alpha) * kr[i];
        float vr   = vcache[(size_t)idx * D_MODEL + d];
        float vrec = qkv[(size_t)t * N_QKV + 2 * D_MODEL + d];
        vf[(size_t)t * D_MODEL + d] = alpha * vrec + (1.f - alpha) * vr;
    }
}

// ---------------------------------------------------------------------------
// Kernel 5: causal ALiBi attention, flash-style, per (head, 64-query group).
// 4 waves x 16 query rows each; K and V (transposed) staged in shared LDS in
// 64-row chunks. Chunk skipping is wave-uniform -> EXEC all-ones at WMMAs.
// LDS: Ks 33.8KB + Vt 34.8KB + Ps 17.4KB = 84KB (<320KB/WGP).
// ---------------------------------------------------------------------------
__global__ __launch_bounds__(128)
void attn_kernel(const float* __restrict__ Q,   // qkv base, stride 6144, q at col 0
                 const float* __restrict__ Kf,  // [T][D]
                 const float* __restrict__ Vf,  // [T][D]
                 float* __restrict__ Oa)        // [T][D]
{
    const int head = blockIdx.x & 15;
    const int qgrp = blockIdx.x >> 4;             // 0..31
    const int tid  = threadIdx.x;
    const int lane = tid & 31;
    const int wave = tid >> 5;
    const int lrow = lane & 15;
    const int lhi  = lane >> 4;

    const int qRow0 = qgrp * 64 + wave * 16;      // this wave's query base
    const int dOff  = head * HD;
    const float slope = exp2f(-0.5f * (float)(head + 1)); // ALiBi (npow==H)

    __shared__ float Ks[64][132];                 // K chunk [kv][dim]
    __shared__ float Vt[128][68];                 // V chunk transposed [dim][kv]
    __shared__ float Ps[4][16][68];               // per-wave scores / probs

    // Q fragments held in registers: qf[i] covers dims i*4 .. i*4+3
    v2f qf[32];
    {
        const float* qp = Q + (size_t)(qRow0 + lrow) * N_QKV + dOff;
        #pragma unroll
        for (int i = 0; i < 32; ++i)
            qf[i] = *(const v2f*)(qp + i * 4 + 2 * lhi);
    }

    v8f o[8] = {};              // 16x128 output accumulator per wave
    float m_run = -__builtin_inff();
    float l_run = 0.f;          // row (lrow) softmax state, duplicated in both halves

    const int kvEnd = qgrp * 64 + 64;
    for (int kv0 = 0; kv0 < kvEnd; kv0 += 64) {
        // ---- stage K chunk + transposed V chunk (all 128 threads) ----
        __syncthreads();
        {
            const int srow = tid >> 1;            // 0..63 (kv row)
            const int c0   = (tid & 1) * 64;      // dim half
            const float* kp = Kf + (size_t)(kv0 + srow) * D_MODEL + dOff + c0;
            const float* vp = Vf + (size_t)(kv0 + srow) * D_MODEL + dOff + c0;
            #pragma unroll
            for (int i = 0; i < 16; ++i) {
                *(float4*)&Ks[srow][c0 + i * 4] = *(const float4*)(kp + i * 4);
                float4 vv = *(const float4*)(vp + i * 4);
                Vt[c0 + i * 4 + 0][srow] = vv.x;
                Vt[c0 + i * 4 + 1][srow] = vv.y;
                Vt[c0 + i * 4 + 2][srow] = vv.z;
                Vt[c0 + i * 4 + 3][srow] = vv.w;
            }
        }
        __syncthreads();

        if (kv0 > qRow0 + 15) continue;   // wave-uniform: fully masked chunk

        // ---- scores S = Q K^T  (4 col tiles of 16 keys) ----
        #pragma unroll
        for (int ct = 0; ct < 4; ++ct) {
            v8f s = {};
            #pragma unroll
            for (int kk = 0; kk < 128; kk += 4) {
                v2f bF = *(const v2f*)&Ks[ct * 16 + lrow][kk + 2 * lhi];
                s = wmma_f32(qf[kk >> 2], bF, s);
            }
            #pragma unroll
            for (int r = 0; r < 8; ++r)
                Ps[wave][r + 8 * lhi][ct * 16 + lrow] = s[r];
        }
        LDS_FENCE();

        // ---- online softmax: 2 lanes per row, lane covers cols lhi*32..+31 ----
        const int gi = qRow0 + lrow;
        float mloc = -__builtin_inff();
        #pragma unroll
        for (int c = 0; c < 32; ++c) {
            int j = kv0 + lhi * 32 + c;
            if (j <= gi) {
                float x = Ps[wave][lrow][lhi * 32 + c] * SM_SCALE
                        + slope * (float)(j - gi);
                mloc = fmaxf(mloc, x);
            }
        }
        float mo    = fmaxf(mloc, __shfl_xor(mloc, 16, 32));
        float m_new = fmaxf(m_run, mo);
        float resc  = (m_run == -__builtin_inff()) ? 0.f : __expf(m_run - m_new);

        float lloc = 0.f;
        #pragma unroll
        for (int c = 0; c < 32; ++c) {
            int j = kv0 + lhi * 32 + c;
            float p = 0.f;
            if (j <= gi) {
                float x = Ps[wave][lrow][lhi * 32 + c] * SM_SCALE
                        + slope * (float)(j - gi);
                p = __expf(x - m_new);
            }
            Ps[wave][lrow][lhi * 32 + c] = p;
            lloc += p;
        }
        lloc += __shfl_xor(lloc, 16, 32);
        l_run = l_run * resc + lloc;
        m_run = m_new;
        LDS_FENCE();

        // ---- rescale running output by per-row factor ----
        float sc[8];
        #pragma unroll
        for (int r = 0; r < 8; ++r) sc[r] = __shfl(resc, r + 8 * lhi, 32);
        #pragma unroll
        for (int nt = 0; nt < 8; ++nt)
            #pragma unroll
            for (int r = 0; r < 8; ++r) o[nt][r] *= sc[r];

        // ---- O += P V  (both operands from LDS: pure ds_load_b64 feeds) ----
        #pragma unroll
        for (int nt = 0; nt < 8; ++nt) {
            #pragma unroll
            for (int kk = 0; kk < 64; kk += 4) {
                v2f aF = *(const v2f*)&Ps[wave][lrow][kk + 2 * lhi];
                v2f bF = *(const v2f*)&Vt[nt * 16 + lrow][kk + 2 * lhi];
                o[nt] = wmma_f32(aF, bF, o[nt]);
            }
        }
        LDS_FENCE();
    }

    // ---- normalize and write ----
    float linv = 1.0f / l_run;
    float li[8];
    #pragma unroll
    for (int r = 0; r < 8; ++r) li[r] = __shfl(linv, r + 8 * lhi, 32);
    #pragma unroll
    for (int nt = 0; nt < 8; ++nt)
        #pragma unroll
        for (int r = 0; r < 8; ++r)
            Oa[(size_t)(qRow0 + r + 8 * lhi) * D_MODEL + dOff + nt * 16 + lrow]
                = o[nt][r] * li[r];
}

// ---------------------------------------------------------------------------
extern "C" void kernel_launch(void* const* d_in, const int* in_sizes, int n_in,
                              void* d_out, int out_size, void* d_ws, size_t ws_size,
                              hipStream_t stream)
{
    (void)in_sizes; (void)n_in; (void)out_size; (void)ws_size;
    const float* hidden    = (const float*)d_in[0];
    const float* Wqkv      = (const float*)d_in[1];
    const float* q_ln_w    = (const float*)d_in[2];
    const float* q_ln_b    = (const float*)d_in[3];
    const float* k_ln_w    = (const float*)d_in[4];
    const float* k_ln_b    = (const float*)d_in[5];
    const float* out_w     = (const float*)d_in[6];
    const float* k_cache   = (const float*)d_in[7];
    const float* v_cache   = (const float*)d_in[8];
    const int*   cache_idx = (const int*)d_in[9];
    const int*   Sk        = (const int*)d_in[10];
    // d_in[11] = layernums (==2 per setup; >=2 path implemented statically)
    float* out = (float*)d_out;

    float* qkv = (float*)d_ws;                               // [T][6144]
    float* kf  = qkv + (size_t)T_TOK * N_QKV;                // [T][2048]
    float* vf  = kf  + (size_t)T_TOK * D_MODEL;              // [T][2048]
    float* att = vf  + (size_t)T_TOK * D_MODEL;              // [T][2048]

    // 1) qkv = clip(hidden @ Wqkv^T)
    gemm_nt_kernel<true><<<dim3(N_QKV / 64, T_TOK / 64), 128, 0, stream>>>(
        hidden, Wqkv, qkv, T_TOK, N_QKV, D_MODEL);

    // 2) layernorm q and k slices in place
    ln_kernel<<<2 * T_TOK, 256, 0, stream>>>(qkv, q_ln_w, q_ln_b, k_ln_w, k_ln_b);

    // 3) cache-reuse gather
    gather_kernel<<<T_TOK, 256, 0, stream>>>(qkv, k_cache, v_cache, cache_idx, kf, vf);

    // 4) alpha blend at Sk rows
    blend_kernel<<<K_SK, 256, 0, stream>>>(qkv, k_cache, v_cache, cache_idx, Sk, kf, vf);

    // 5) causal ALiBi attention
    attn_kernel<<<H_HEADS * (T_TOK / 64), 128, 0, stream>>>(qkv, kf, vf, att);

    // 6) out = att @ out_w^T
    gemm_nt_kernel<false><<<dim3(D_MODEL / 64, T_TOK / 64), 128, 0, stream>>>(
        att, out_w, out, T_TOK, D_MODEL, D_MODEL);
}